// ACIE_Core_88347477278870
// MI455X (gfx1250) — compile-verified
//
#include <hip/hip_runtime.h>
#include <math.h>

typedef __attribute__((ext_vector_type(2))) float v2f;
typedef __attribute__((ext_vector_type(8))) float v8f;
typedef __attribute__((ext_vector_type(4))) int i4;

#define BATCH 8192
#define INPUT_DIM 10240
#define NODES 1024
#define ACTIONS 64
#define HIDDEN 128
#define BINS 100

#define MODE_NONE      0
#define MODE_ROWSCALE  1
#define MODE_BIAS      2
#define MODE_BIAS_RELU 3
#define MODE_SIGNPERT  4

// ---- CDNA5 async global->LDS path (ASYNCcnt), probe-guarded ---------------
#if defined(__gfx1250__) &&                                             \
    __has_builtin(__builtin_amdgcn_global_load_async_to_lds_b128) &&    \
    __has_builtin(__builtin_amdgcn_global_load_async_to_lds_b32) &&     \
    __has_builtin(__builtin_amdgcn_s_wait_asynccnt)
#define USE_ASYNC_LDS 1
#else
#define USE_ASYNC_LDS 0
#endif

#if USE_ASYNC_LDS
typedef i4  __attribute__((address_space(1))) *g_i4p;
typedef i4  __attribute__((address_space(3))) *l_i4p;
typedef int __attribute__((address_space(1))) *g_i1p;
typedef int __attribute__((address_space(3))) *l_i1p;

__device__ __forceinline__ void async_cp_b128(const float* g, float* l) {
  __builtin_amdgcn_global_load_async_to_lds_b128(
      (g_i4p)(i4*)g, (l_i4p)(i4*)l, 0, 0);
}
__device__ __forceinline__ void async_cp_b32(const float* g, float* l) {
  __builtin_amdgcn_global_load_async_to_lds_b32(
      (g_i1p)(int*)g, (l_i1p)(int*)l, 0, 0);
}
__device__ __forceinline__ void async_wait0() {
  __builtin_amdgcn_s_wait_asynccnt(0);
}
#endif

// ---------------------------------------------------------------------------
// Layer 1: per-row histogram + Shannon entropy gate -> keep[b] in {0,1}
// ---------------------------------------------------------------------------
__global__ __launch_bounds__(256) void entropy_keep_kernel(
    const float* __restrict__ es, float* __restrict__ keep) {
  __shared__ int hist[BINS];
  const int row = blockIdx.x;
  for (int i = threadIdx.x; i < BINS; i += 256) hist[i] = 0;
  __syncthreads();
  const float* x = es + (size_t)row * INPUT_DIM;
  for (int i = threadIdx.x; i < INPUT_DIM; i += 256) {
    float v = x[i];
    if (v >= -5.0f && v <= 5.0f) {
      int idx = (int)floorf((v + 5.0f) * 10.0f);
      idx = idx < 0 ? 0 : (idx > BINS - 1 ? BINS - 1 : idx);
      atomicAdd(&hist[idx], 1);
    }
  }
  __syncthreads();
  if (threadIdx.x == 0) {
    float tot = 0.0f;
    for (int i = 0; i < BINS; ++i) tot += (float)hist[i];
    float inv = 1.0f / (tot + 1e-9f);
    float ent = 0.0f;
    for (int i = 0; i < BINS; ++i) {
      float p = (float)hist[i] * inv;
      ent -= p * logf(p + 1e-9f);
    }
    keep[row] = (ent >= 2.5f) ? 1.0f : 0.0f;
  }
}

// ---------------------------------------------------------------------------
// fp32 WMMA GEMM, block tile 128x128, 8 waves of 32x64 (2x4 WMMA subtiles).
// Double-buffered LDS with CDNA5 async global->LDS staging when available.
// A-row stride 20 floats: 80B (16B aligned for async B128) and conflict-free
// fragment reads (20*m mod 64 distinct for m=0..15).  B-row stride 132 floats.
// ---------------------------------------------------------------------------
__device__ __forceinline__ void wmma_chunk(
    const float (*__restrict__ As)[20], const float (*__restrict__ Bs)[132],
    int warow, int wacol, int lane16, int khalf, v8f acc[2][4]) {
#pragma unroll
  for (int kk = 0; kk < 16; kk += 4) {
    v2f a[2], b[4];
#pragma unroll
    for (int mi = 0; mi < 2; ++mi) {
      int m = warow + mi * 16 + lane16;
      a[mi].x = As[m][kk + khalf];
      a[mi].y = As[m][kk + khalf + 1];
    }
#pragma unroll
    for (int ni = 0; ni < 4; ++ni) {
      int n = wacol + ni * 16 + lane16;
      b[ni].x = Bs[kk + khalf][n];
      b[ni].y = Bs[kk + khalf + 1][n];
    }
#pragma unroll
    for (int mi = 0; mi < 2; ++mi)
#pragma unroll
      for (int ni = 0; ni < 4; ++ni)
        acc[mi][ni] = __builtin_amdgcn_wmma_f32_16x16x4_f32(
            false, a[mi], false, b[ni], (short)0, acc[mi][ni], false, false);
  }
}

__global__ __launch_bounds__(256) void gemm_f32_wmma(
    const float* __restrict__ A, int lda,
    const float* __restrict__ B, int ldb, int bNT,
    float* __restrict__ C, int ldc,
    int M, int N, int K, int mode,
    const float* __restrict__ rowscale,   // MODE_ROWSCALE
    const float* __restrict__ bias,       // MODE_BIAS / MODE_BIAS_RELU
    const float* __restrict__ aux)        // MODE_SIGNPERT: state (ld = ldc)
{
  __shared__ float As[2][128][20];
  __shared__ float Bs[2][16][132];

  const int tid    = threadIdx.x;
  const int wave   = tid >> 5;
  const int lane   = tid & 31;
  const int lane16 = lane & 15;
  const int khalf  = (lane < 16) ? 0 : 2;
  const int warow  = (wave >> 1) * 32;   // 4 row groups
  const int wacol  = (wave & 1) * 64;    // 2 col groups
  const int m0     = blockIdx.y * 128;
  const int n0     = blockIdx.x * 128;

  v8f zero = {};
  v8f acc[2][4];
#pragma unroll
  for (int i = 0; i < 2; ++i)
#pragma unroll
    for (int j = 0; j < 4; ++j) acc[i][j] = zero;

  const int nchunks = K >> 4;

#if USE_ASYNC_LDS
  // -------- async double-buffered pipeline --------
  auto stage = [&](int buf, int k0) {
#pragma unroll
    for (int t = 0; t < 2; ++t) {                  // A: 128x16, 2 float4/thr
      int i = tid + t * 256;
      int r = i >> 2, c = (i & 3) << 2;
      async_cp_b128(A + (size_t)(m0 + r) * lda + k0 + c, &As[buf][r][c]);
    }
    if (bNT) {                                     // B [N,K]: transpose stage
#pragma unroll
      for (int t = 0; t < 2; ++t) {
        int i = tid + t * 256;
        int n = i >> 2, c = (i & 3) << 2;
        const float* g = B + (size_t)(n0 + n) * ldb + k0 + c;
#pragma unroll
        for (int j = 0; j < 4; ++j) async_cp_b32(g + j, &Bs[buf][c + j][n]);
      }
    } else {                                       // B [K,N]: direct stage
#pragma unroll
      for (int t = 0; t < 2; ++t) {
        int i = tid + t * 256;
        int kr = i >> 5, c = (i & 31) << 2;
        async_cp_b128(B + (size_t)(k0 + kr) * ldb + n0 + c, &Bs[buf][kr][c]);
      }
    }
  };

  stage(0, 0);
  async_wait0();
  __syncthreads();
  for (int c = 0; c < nchunks; ++c) {
    const int cur = c & 1;
    if (c + 1 < nchunks) stage(cur ^ 1, (c + 1) << 4);  // overlap with WMMA
    wmma_chunk(As[cur], Bs[cur], warow, wacol, lane16, khalf, acc);
    if (c + 1 < nchunks) {
      async_wait0();
      __syncthreads();
    }
  }
#else
  // -------- sync single-buffer fallback --------
  for (int c = 0; c < nchunks; ++c) {
    const int k0 = c << 4;
#pragma unroll
    for (int t = 0; t < 2; ++t) {
      int i = tid + t * 256;
      int r = i >> 2, cc = (i & 3) << 2;
      const float4 v =
          *reinterpret_cast<const float4*>(A + (size_t)(m0 + r) * lda + k0 + cc);
      As[0][r][cc + 0] = v.x; As[0][r][cc + 1] = v.y;
      As[0][r][cc + 2] = v.z; As[0][r][cc + 3] = v.w;
    }
    if (bNT) {
#pragma unroll
      for (int t = 0; t < 2; ++t) {
        int i = tid + t * 256;
        int n = i >> 2, cc = (i & 3) << 2;
        const float4 v =
            *reinterpret_cast<const float4*>(B + (size_t)(n0 + n) * ldb + k0 + cc);
        Bs[0][cc + 0][n] = v.x; Bs[0][cc + 1][n] = v.y;
        Bs[0][cc + 2][n] = v.z; Bs[0][cc + 3][n] = v.w;
      }
    } else {
#pragma unroll
      for (int t = 0; t < 2; ++t) {
        int i = tid + t * 256;
        int kr = i >> 5, cc = (i & 31) << 2;
        const float4 v =
            *reinterpret_cast<const float4*>(B + (size_t)(k0 + kr) * ldb + n0 + cc);
        Bs[0][kr][cc + 0] = v.x; Bs[0][kr][cc + 1] = v.y;
        Bs[0][kr][cc + 2] = v.z; Bs[0][kr][cc + 3] = v.w;
      }
    }
    __syncthreads();
    wmma_chunk(As[0], Bs[0], warow, wacol, lane16, khalf, acc);
    __syncthreads();
  }
#endif

  // -------- epilogue --------
#pragma unroll
  for (int mi = 0; mi < 2; ++mi)
#pragma unroll
    for (int ni = 0; ni < 4; ++ni)
#pragma unroll
      for (int r = 0; r < 8; ++r) {
        int row = m0 + warow + mi * 16 + ((lane < 16) ? r : r + 8);
        int col = n0 + wacol + ni * 16 + lane16;
        float v = acc[mi][ni][r];
        if (mode == MODE_ROWSCALE) {
          v *= rowscale[row];
        } else if (mode == MODE_BIAS) {
          v += bias[col];
        } else if (mode == MODE_BIAS_RELU) {
          v += bias[col];
          v = v > 0.0f ? v : 0.0f;
        } else if (mode == MODE_SIGNPERT) {
          float sg = (v > 0.0f) ? 1.0f : ((v < 0.0f) ? -1.0f : 0.0f);
          v = aux[(size_t)row * ldc + col] - 0.1f * sg;
        }
        C[(size_t)row * ldc + col] = v;
      }
}

// ---------------------------------------------------------------------------
// c[h] = (sum_a W2[a,h]) / (B*A)   -- analytic grad of mean(logits) wrt h
// ---------------------------------------------------------------------------
__global__ void colsum_w2_kernel(const float* __restrict__ W2,
                                 float* __restrict__ c) {
  int h = threadIdx.x;  // 128 threads
  float s = 0.0f;
  for (int a = 0; a < ACTIONS; ++a) s += W2[a * HIDDEN + h];
  c[h] = s * (1.0f / ((float)BATCH * (float)ACTIONS));
}

// g[b,h] = relu'(pre1[b,h]) * c[h]
__global__ void gmask_kernel(const float* __restrict__ pre1,
                             const float* __restrict__ c,
                             float* __restrict__ g, int n) {
  int i = blockIdx.x * 256 + threadIdx.x;
  if (i < n) g[i] = (pre1[i] > 0.0f) ? c[i & (HIDDEN - 1)] : 0.0f;
}

// ---------------------------------------------------------------------------
// logits = h2 @ W2^T + b2, then row softmax.  One wave32 per batch row.
// ---------------------------------------------------------------------------
__global__ __launch_bounds__(256) void policy_softmax_kernel(
    const float* __restrict__ h2, const float* __restrict__ W2,
    const float* __restrict__ b2, float* __restrict__ out) {
  __shared__ float W2s[ACTIONS * HIDDEN];
  __shared__ float b2s[ACTIONS];
  __shared__ float hs[8][HIDDEN];
  const int tid = threadIdx.x;
  for (int i = tid; i < ACTIONS * HIDDEN; i += 256) W2s[i] = W2[i];
  if (tid < ACTIONS) b2s[tid] = b2[tid];
  const int wave = tid >> 5, lane = tid & 31;
  const int row = blockIdx.x * 8 + wave;
  const float* h = h2 + (size_t)row * HIDDEN;
#pragma unroll
  for (int j = 0; j < 4; ++j) hs[wave][lane * 4 + j] = h[lane * 4 + j];
  __syncthreads();

  float v0 = b2s[lane], v1 = b2s[lane + 32];
  for (int k = 0; k < HIDDEN; ++k) {
    float hv = hs[wave][k];
    v0 = fmaf(hv, W2s[lane * HIDDEN + k], v0);
    v1 = fmaf(hv, W2s[(lane + 32) * HIDDEN + k], v1);
  }
  float m = fmaxf(v0, v1);
  for (int off = 16; off >= 1; off >>= 1) m = fmaxf(m, __shfl_xor(m, off, 32));
  float e0 = expf(v0 - m), e1 = expf(v1 - m);
  float s = e0 + e1;
  for (int off = 16; off >= 1; off >>= 1) s += __shfl_xor(s, off, 32);
  float inv = 1.0f / s;
  out[(size_t)row * ACTIONS + lane]      = e0 * inv;
  out[(size_t)row * ACTIONS + 32 + lane] = e1 * inv;
}

// ---------------------------------------------------------------------------
// semantic_loss = sum(|adj| * mask); two-stage deterministic reduction
// ---------------------------------------------------------------------------
__global__ __launch_bounds__(256) void semloss_partial_kernel(
    const float* __restrict__ adj, const float* __restrict__ msk,
    float* __restrict__ part) {
  __shared__ float red[256];
  const int row = blockIdx.x;
  float s = 0.0f;
  for (int i = threadIdx.x; i < NODES; i += 256) {
    size_t idx = (size_t)row * NODES + i;
    s += fabsf(adj[idx]) * msk[idx];
  }
  red[threadIdx.x] = s;
  __syncthreads();
  for (int off = 128; off > 0; off >>= 1) {
    if (threadIdx.x < off) red[threadIdx.x] += red[threadIdx.x + off];
    __syncthreads();
  }
  if (threadIdx.x == 0) part[row] = red[0];
}

__global__ __launch_bounds__(256) void semloss_final_kernel(
    const float* __restrict__ part, float* __restrict__ out) {
  __shared__ float red[256];
  float s = 0.0f;
  for (int i = threadIdx.x; i < NODES; i += 256) s += part[i];
  red[threadIdx.x] = s;
  __syncthreads();
  for (int off = 128; off > 0; off >>= 1) {
    if (threadIdx.x < off) red[threadIdx.x] += red[threadIdx.x + off];
    __syncthreads();
  }
  if (threadIdx.x == 0) out[0] = red[0];
}

// ---------------------------------------------------------------------------
extern "C" void kernel_launch(void* const* d_in, const int* in_sizes, int n_in,
                              void* d_out, int out_size, void* d_ws,
                              size_t ws_size, hipStream_t stream) {
  const float* E   = (const float*)d_in[0];  // [8192,10240]
  const float* S   = (const float*)d_in[1];  // [1024,10240]
  const float* Adj = (const float*)d_in[2];  // [1024,1024]
  const float* Msk = (const float*)d_in[3];  // [1024,1024]
  const float* W1  = (const float*)d_in[4];  // [128,1024]
  const float* b1  = (const float*)d_in[5];  // [128]
  const float* W2  = (const float*)d_in[6];  // [64,128]
  const float* b2  = (const float*)d_in[7];  // [64]
  float* out = (float*)d_out;
  float* ws  = (float*)d_ws;

  float* keep = ws;                                  // 8192
  float* cvec = ws + 8192;                           // 128
  float* part = ws + 8448;                           // 1024
  float* buf1 = ws + 16384;                          // filtered, later s2
  float* buf2 = buf1 + (size_t)BATCH * NODES;        // state
  float* buf3 = buf2 + (size_t)BATCH * NODES;        // pre1, later h2
  float* buf4 = buf3 + (size_t)BATCH * HIDDEN;       // g

  // 1) entropy gate
  entropy_keep_kernel<<<BATCH, 256, 0, stream>>>(E, keep);
  // 2) filtered = (E @ S^T) * keep[row]   (M=8192,N=1024,K=10240, NT)
  gemm_f32_wmma<<<dim3(NODES / 128, BATCH / 128), 256, 0, stream>>>(
      E, INPUT_DIM, S, INPUT_DIM, 1, buf1, NODES, BATCH, NODES, INPUT_DIM,
      MODE_ROWSCALE, keep, nullptr, nullptr);
  // 3) state = filtered @ adjacency       (NN)
  gemm_f32_wmma<<<dim3(NODES / 128, BATCH / 128), 256, 0, stream>>>(
      buf1, NODES, Adj, NODES, 0, buf2, NODES, BATCH, NODES, NODES,
      MODE_NONE, nullptr, nullptr, nullptr);
  // 4) pre1 = state @ W1^T + b1           (NT)
  gemm_f32_wmma<<<dim3(HIDDEN / 128, BATCH / 128), 256, 0, stream>>>(
      buf2, NODES, W1, NODES, 1, buf3, HIDDEN, BATCH, HIDDEN, NODES,
      MODE_BIAS, nullptr, b1, nullptr);
  // 5) c = colsum(W2)/(B*A);  6) g = relu'(pre1)*c
  colsum_w2_kernel<<<1, HIDDEN, 0, stream>>>(W2, cvec);
  gmask_kernel<<<(BATCH * HIDDEN) / 256, 256, 0, stream>>>(
      buf3, cvec, buf4, BATCH * HIDDEN);
  // 7) s2 = state - 0.1*sign(g @ W1)      (NN, fused FGSM epilogue)
  gemm_f32_wmma<<<dim3(NODES / 128, BATCH / 128), 256, 0, stream>>>(
      buf4, HIDDEN, W1, NODES, 0, buf1, NODES, BATCH, NODES, HIDDEN,
      MODE_SIGNPERT, nullptr, nullptr, buf2);
  // 8) h2 = relu(s2 @ W1^T + b1)          (NT)
  gemm_f32_wmma<<<dim3(HIDDEN / 128, BATCH / 128), 256, 0, stream>>>(
      buf1, NODES, W1, NODES, 1, buf3, HIDDEN, BATCH, HIDDEN, NODES,
      MODE_BIAS_RELU, nullptr, b1, nullptr);
  // 9) action_probs = softmax(h2 @ W2^T + b2)
  policy_softmax_kernel<<<BATCH / 8, 256, 0, stream>>>(buf3, W2, b2, out);
  // 10) semantic loss + adjacency pass-through
  semloss_partial_kernel<<<NODES, 256, 0, stream>>>(Adj, Msk, part);
  semloss_final_kernel<<<1, 256, 0, stream>>>(
      part, out + (size_t)BATCH * ACTIONS + (size_t)NODES * NODES);
  (void)hipMemcpyAsync(out + (size_t)BATCH * ACTIONS, Adj,
                       (size_t)NODES * NODES * sizeof(float),
                       hipMemcpyDeviceToDevice, stream);
}